// RiemannianManifold_54752243090098
// MI455X (gfx1250) — compile-verified
//
#include <hip/hip_runtime.h>
#include <math.h>

typedef float v2f __attribute__((ext_vector_type(2)));
typedef float v8f __attribute__((ext_vector_type(8)));

constexpr int D  = 32;
constexpr int H  = 64;
constexpr int Bb = 2;
constexpr int Nn = 512;
constexpr int Mm = 512;

__device__ __forceinline__ float silu_f(float x) {
    return x / (1.0f + expf(-x));
}
__device__ __forceinline__ float softplus_f(float x) {
    // stable log(1+exp(x)) = max(x,0) + log1p(exp(-|x|))
    return fmaxf(x, 0.0f) + log1pf(expf(-fabsf(x)));
}

// ---------------------------------------------------------------------------
// Kernel 1: per-token Cholesky-factor MLP.  One block per token (2048 blocks).
//   h = silu(x@w1+b1)   [64]
//   raw = h@w2+b2       [1024], soft-clamped, +I, tril, softplus diag
// ---------------------------------------------------------------------------
__global__ __launch_bounds__(256) void rm_factor_kernel(
    const float* __restrict__ Q, const float* __restrict__ K,
    const float* __restrict__ w1, const float* __restrict__ b1,
    const float* __restrict__ w2, const float* __restrict__ b2,
    float* __restrict__ Lws)
{
    __shared__ float xs[D];
    __shared__ float hs[H];
    const int t   = blockIdx.x;
    const int tid = threadIdx.x;
    const float* x = (t < Bb * Nn) ? (Q + (size_t)t * D)
                                   : (K + (size_t)(t - Bb * Nn) * D);
    if (tid < D) xs[tid] = x[tid];
    __syncthreads();
    if (tid < H) {
        float acc = b1[tid];
        #pragma unroll
        for (int i = 0; i < D; ++i) acc += xs[i] * w1[i * H + tid];
        hs[tid] = silu_f(acc);
    }
    __syncthreads();
    float* Lt = Lws + (size_t)t * (D * D);
    #pragma unroll
    for (int r = 0; r < 4; ++r) {
        const int p = tid + 256 * r;
        float acc = b2[p];
        #pragma unroll 8
        for (int j = 0; j < H; ++j) acc += hs[j] * w2[j * (D * D) + p];
        const float v = 5.0f * tanhf(acc * 0.2f);
        const int row = p >> 5, col = p & 31;
        float o;
        if (col > row)       o = 0.0f;
        else if (col == row) o = softplus_f(v + 1.0f) + 1e-4f;
        else                 o = v;
        Lt[p] = o;
    }
}

// ---------------------------------------------------------------------------
// Kernel 2 (two passes): per center token c (owning factor L), GEMM
//   (Rows_b - 1*center^T) @ L   of shape 512x32x32 via V_WMMA_F32_16X16X4_F32,
// then row-wise squared norm.
//   pass FINAL=false: centers are Q rows (vq term), rows are K  -> out = 0.5*s
//   pass FINAL=true : centers are K rows (vk term), rows are Q  -> out = sqrt(clip(out + 0.5*s))
// WMMA 16x16x4 f32 fragment layout (ISA 7.12.2):
//   A: lanes 0-15 hold row M=lane, v0/v1 = K,K+1; lanes 16-31 = K+2,K+3
//   B: lane = output column (mod 16), v0/v1 = K rows (same half split)
//   C/D: vgpr v holds row v (lanes 0-15) / row v+8 (lanes 16-31), col = lane&15
// ---------------------------------------------------------------------------
template <bool FINAL>
__global__ __launch_bounds__(256) void rm_pair_kernel(
    const float* __restrict__ Rows,   // [B, R, D]   (K for vq pass, Q for vk pass)
    const float* __restrict__ Cen,    // [B*C, D]    center tokens owning L
    const float* __restrict__ Lfac,   // [B*C, D*D]
    float* __restrict__ out,          // [B, N, M]
    int rowsPerBatch)
{
    __shared__ float Ls[D * D];
    __shared__ float cs[D];
    const int bc  = blockIdx.x;       // b*512 + c   (C == 512 in both passes)
    const int b   = bc >> 9;
    const int tid = threadIdx.x;

    const float* Lg = Lfac + (size_t)bc * (D * D);
    #pragma unroll
    for (int r = 0; r < 4; ++r) Ls[tid + 256 * r] = Lg[tid + 256 * r];
    if (tid < D) cs[tid] = Cen[(size_t)bc * D + tid];
    __syncthreads();

    const float* Rb = Rows + (size_t)b * rowsPerBatch * D;
    const int wave = tid >> 5;
    const int lane = tid & 31;
    const int half = lane >> 4;
    const int l15  = lane & 15;
    const int koff = half * 2;

    for (int s = wave; s < rowsPerBatch / 16; s += 8) {
        const int m0   = s * 16;
        const int arow = m0 + l15;
        const float* ar = Rb + (size_t)arow * D;

        v8f c0 = {}; v8f c1 = {};
        #pragma unroll
        for (int kk = 0; kk < 8; ++kk) {
            const int k = kk * 4 + koff;
            v2f a;   a.x   = ar[k]     - cs[k];
                     a.y   = ar[k + 1] - cs[k + 1];
            v2f bL;  bL.x  = Ls[k * D + l15];        bL.y  = Ls[(k + 1) * D + l15];
            v2f bH;  bH.x  = Ls[k * D + 16 + l15];   bH.y  = Ls[(k + 1) * D + 16 + l15];
            c0 = __builtin_amdgcn_wmma_f32_16x16x4_f32(false, a, false, bL,
                                                       (short)0, c0, false, false);
            c1 = __builtin_amdgcn_wmma_f32_16x16x4_f32(false, a, false, bH,
                                                       (short)0, c1, false, false);
        }

        // Row-wise sum of squares over the 32 output columns.
        float sums[8];
        #pragma unroll
        for (int v = 0; v < 8; ++v) {
            float p = c0[v] * c0[v] + c1[v] * c1[v];
            p += __shfl_xor(p, 1, 32);
            p += __shfl_xor(p, 2, 32);
            p += __shfl_xor(p, 4, 32);
            p += __shfl_xor(p, 8, 32);
            sums[v] = p;
        }

        if (l15 == 0) {
            const int rbase = m0 + half * 8;
            if (!FINAL) {
                // vq pass: bc = b*N + n, row index is m -> contiguous out row
                float* o = out + (size_t)bc * Mm + rbase;
                #pragma unroll
                for (int v = 0; v < 8; ++v) o[v] = 0.5f * sums[v];
            } else {
                // vk pass: bc = b*M + m, row index is n -> strided column RMW
                const int m = bc & (Mm - 1);
                #pragma unroll
                for (int v = 0; v < 8; ++v) {
                    const int n = rbase + v;
                    float* o = out + ((size_t)b * Nn + n) * Mm + m;
                    float d2 = *o + 0.5f * sums[v];
                    d2 = fminf(fmaxf(d2, 1e-6f), 1e6f);
                    *o = sqrtf(d2);
                }
            }
        }
    }
}

extern "C" void kernel_launch(void* const* d_in, const int* in_sizes, int n_in,
                              void* d_out, int out_size, void* d_ws, size_t ws_size,
                              hipStream_t stream) {
    (void)in_sizes; (void)n_in; (void)out_size; (void)ws_size;
    const float* Q  = (const float*)d_in[0];
    const float* K  = (const float*)d_in[1];
    const float* w1 = (const float*)d_in[2];
    const float* b1 = (const float*)d_in[3];
    const float* w2 = (const float*)d_in[4];
    const float* b2 = (const float*)d_in[5];
    float* out = (float*)d_out;

    // Workspace: Lq [B*N*1024 f32] followed by Lk [B*M*1024 f32] = 8 MiB total.
    float* Lq = (float*)d_ws;
    float* Lk = Lq + (size_t)Bb * Nn * D * D;

    rm_factor_kernel<<<Bb * (Nn + Mm), 256, 0, stream>>>(Q, K, w1, b1, w2, b2, Lq);
    rm_pair_kernel<false><<<Bb * Nn, 256, 0, stream>>>(K, Q, Lq, out, Mm);
    rm_pair_kernel<true ><<<Bb * Mm, 256, 0, stream>>>(Q, K, Lk, out, Nn);
}